// DigitConvolutionalModel_33964601377027
// MI455X (gfx1250) — compile-verified
//
#include <hip/hip_runtime.h>
#include <stdint.h>

typedef __attribute__((ext_vector_type(2))) float v2f;
typedef __attribute__((ext_vector_type(8))) float v8f;

#define IMG   28
#define OUTW  26
#define FEAT  676          // 26*26
#define KPAD  678          // even row stride: 16 strided lanes -> 16 distinct banks
#define ROWS  32           // samples per block (2 waves x 16-row WMMA tiles)
#define NT    64           // 2 wave32 waves

__global__ __launch_bounds__(NT)
void conv_wmma_kernel(const float* __restrict__ x,
                      const float* __restrict__ cw,
                      const float* __restrict__ W,
                      const float* __restrict__ bias,
                      float* __restrict__ out)
{
    extern __shared__ float lds[];
    float* Wt   = lds;               // [16][KPAD] W transposed (LDS offset 0)
    float* feat = lds + 16 * KPAD;   // [ROWS][KPAD] conv features

    const int tid  = threadIdx.x;
    const int base = blockIdx.x * ROWS;

    // ---- Phase 0a: async gather-transpose W (676x10) -> Wt[n][k].
    // GLOBAL_LOAD_ASYNC_TO_LDS_B32 is per-lane: lane supplies global addr and
    // LDS dest. Tracked by ASYNCcnt; overlaps with the conv phase below.
    // Dynamic LDS starts at offset 0 (no static __shared__), so the LDS
    // address operand is the plain byte offset of Wt[n*KPAD + k].
    for (int n = 0; n < 10; ++n) {
        for (int k = tid; k < FEAT; k += NT) {
            const uint32_t laddr = (uint32_t)(n * KPAD + k) * 4u;
            const uint64_t gaddr = (uint64_t)(uintptr_t)(W + k * 10 + n);
            asm volatile("global_load_async_to_lds_b32 %0, %1, off"
                         :: "v"(laddr), "v"(gaddr) : "memory");
        }
    }
    // Zero-pad columns n = 10..15 (plain DS stores, small).
    for (int n = 10; n < 16; ++n)
        for (int k = tid; k < FEAT; k += NT)
            Wt[n * KPAD + k] = 0.0f;

    // 3x3 kernel weights -> registers (uniform address -> scalar loads)
    const float k00 = cw[0], k01 = cw[1], k02 = cw[2];
    const float k10 = cw[3], k11 = cw[4], k12 = cw[5];
    const float k20 = cw[6], k21 = cw[7], k22 = cw[8];

    // ---- Phase 1: 3x3 VALID conv + ReLU into feat.
    // Outer over feature index (11 iters -> only 11 div/mod per thread),
    // inner over the 32 images with pure pointer increments.
    for (int f = tid; f < FEAT; f += NT) {
        const int r = f / OUTW;
        const int c = f - r * OUTW;
        const float* p = x + (size_t)base * 784 + r * IMG + c;
        float* frow = feat + f;
#pragma unroll 4
        for (int img = 0; img < ROWS; ++img) {
            __builtin_prefetch(p + 8 * 784, 0, 1);   // stream-ahead for HBM x
            float s = p[ 0] * k00 + p[ 1] * k01 + p[ 2] * k02
                    + p[28] * k10 + p[29] * k11 + p[30] * k12
                    + p[56] * k20 + p[57] * k21 + p[58] * k22;
            frow[img * KPAD] = s > 0.0f ? s : 0.0f;
            p += 784;
        }
    }

    // Async W-transpose must have landed before the GEMM reads Wt.
    asm volatile("s_wait_asynccnt 0x0" ::: "memory");
    __syncthreads();

    // ---- Phase 2: per-wave 16x16 tile, D = feat(16x676) @ Wt^T + bias ----
    // V_WMMA_F32_16X16X4_F32: A 16x4, lanes 0-15 hold K=0,1; lanes 16-31 K=2,3.
    const int lane = tid & 31;
    const int wv   = tid >> 5;          // wave id: which 16-row tile
    const int m    = lane & 15;         // row (A) / col (B) index
    const int half = lane >> 4;         // K sub-offset selector

    const float* Arow = feat + (wv * 16 + m) * KPAD + 2 * half;
    const float* Bcol = Wt   + m * KPAD + 2 * half;

    v8f acc = {};
#pragma unroll 13                        // 169 = 13 * 13 k-steps
    for (int k = 0; k < FEAT; k += 4) {
        const v2f a = *(const v2f*)(Arow + k);   // 8B aligned (KPAD even, k%4==0)
        const v2f b = *(const v2f*)(Bcol + k);
        acc = __builtin_amdgcn_wmma_f32_16x16x4_f32(
            /*neg_a=*/false, a, /*neg_b=*/false, b,
            /*c_mod=*/(short)0, acc, /*reuse_a=*/false, /*reuse_b=*/false);
    }

    // D layout: VGPR v -> M = v + 8*half ; N = lane & 15
    const int n = lane & 15;
    if (n < 10) {
        const float bv = bias[n];
        float* orow = out + (size_t)(base + wv * 16 + half * 8) * 10 + n;
#pragma unroll
        for (int v = 0; v < 8; ++v)
            orow[v * 10] = acc[v] + bv;
    }
}

extern "C" void kernel_launch(void* const* d_in, const int* in_sizes, int n_in,
                              void* d_out, int out_size, void* d_ws, size_t ws_size,
                              hipStream_t stream) {
    const float* x  = (const float*)d_in[0];   // (B, 784) fp32
    const float* cw = (const float*)d_in[1];   // (3, 3)   fp32
    const float* W  = (const float*)d_in[2];   // (676,10) fp32
    const float* b  = (const float*)d_in[3];   // (10,)    fp32
    float* out = (float*)d_out;                // (B, 10)  fp32

    const int B = in_sizes[0] / 784;           // 32768
    const size_t shmem = (size_t)(16 + ROWS) * KPAD * sizeof(float); // ~130 KB

    conv_wmma_kernel<<<B / ROWS, NT, shmem, stream>>>(x, cw, W, b, out);
}